// ECELoss_88493506167218
// MI455X (gfx1250) — compile-verified
//
#include <hip/hip_runtime.h>

typedef __attribute__((ext_vector_type(2))) float v2f;
typedef __attribute__((ext_vector_type(8))) float v8f;

#define NBINS   15
#define HROWS   16   // 15 real bins + 1 zero row so WMMA sees a clean 16x32 tile
#define HSTRIDE 32   // one slot per lane -> conflict-free LDS banking

// Column-sum of a 16x32 fp32 LDS tile via chained V_WMMA_F32_16X16X4_F32.
// A = ones(16x4); B_g[k][n] = tile[n][4g+k]. After 8 accumulating WMMAs,
// D[0][n] (VGPR0 of lanes 0..15) = sum over all 32 columns of row n.
// Must be called by one fully-active wave (EXEC all 1s).
__device__ __forceinline__ v8f lds_colsum_16x32(const float* __restrict__ s, int lane) {
  v8f c = {};
  const int bin  = lane & 15;
  const int koff = (lane >> 4) << 1;  // lanes 0-15 hold K=0,1; lanes 16-31 hold K=2,3
  v2f a; a[0] = 1.0f; a[1] = 1.0f;    // A-matrix: all ones (16x4)
#pragma unroll
  for (int g = 0; g < 8; ++g) {
    v2f b;
    b[0] = s[bin * HSTRIDE + 4 * g + koff + 0];
    b[1] = s[bin * HSTRIDE + 4 * g + koff + 1];
    c = __builtin_amdgcn_wmma_f32_16x16x4_f32(
        /*neg_a=*/false, a, /*neg_b=*/false, b,
        /*c_mod=*/(short)0, c, /*reuse_a=*/false, /*reuse_b=*/false);
  }
  return c;
}

__device__ __forceinline__ int bin_of(float conf) {
  // torch ECE bins: i/15 < conf <= (i+1)/15  ->  ceil(conf*15)-1, clamped
  int idx = (int)ceilf(conf * 15.0f) - 1;
  idx = idx < 0 ? 0 : idx;
  idx = idx > NBINS - 1 ? NBINS - 1 : idx;
  return idx;
}

__global__ void __launch_bounds__(64) ece_zero_ws(float* __restrict__ ws) {
  int i = threadIdx.x;
  if (i < 48) ws[i] = 0.0f;
}

// ws layout: [0..15] counts, [16..31] sum_conf, [32..47] sum_acc
__global__ void __launch_bounds__(256)
ece_hist(const float* __restrict__ logits, const float* __restrict__ labels,
         float* __restrict__ ws, int n) {
  __shared__ float s_conf[HROWS * HSTRIDE];
  __shared__ float s_cnt [HROWS * HSTRIDE];
  __shared__ float s_acc [HROWS * HSTRIDE];

  const int tid  = threadIdx.x;
  const int lane = tid & 31;
  for (int i = tid; i < HROWS * HSTRIDE; i += blockDim.x) {
    s_conf[i] = 0.0f; s_cnt[i] = 0.0f; s_acc[i] = 0.0f;
  }
  __syncthreads();

  const int gtid = blockIdx.x * blockDim.x + tid;
  const int nthr = gridDim.x * blockDim.x;
  const int n4   = n & ~3;

  // Streaming main loop: 16B-per-lane loads of both tensors (HBM-bound path).
  for (int i = gtid * 4; i < n4; i += nthr * 4) {
    const float4 lg = *reinterpret_cast<const float4*>(logits + i);
    const float4 lb = *reinterpret_cast<const float4*>(labels + i);
    const float xs[4] = {lg.x, lg.y, lg.z, lg.w};
    const float ys[4] = {lb.x, lb.y, lb.z, lb.w};
#pragma unroll
    for (int e = 0; e < 4; ++e) {
      const float conf = __builtin_amdgcn_rcpf(1.0f + __expf(-xs[e]));  // sigmoid
      const int slot = bin_of(conf) * HSTRIDE + lane;  // bank == lane: conflict-free
      atomicAdd(&s_conf[slot], conf);
      atomicAdd(&s_cnt [slot], 1.0f);
      atomicAdd(&s_acc [slot], ys[e]);
    }
  }
  // Scalar tail (n is 64M in practice, so this is empty; kept for generality).
  for (int i = n4 + gtid; i < n; i += nthr) {
    const float conf = __builtin_amdgcn_rcpf(1.0f + __expf(-logits[i]));
    const int slot = bin_of(conf) * HSTRIDE + lane;
    atomicAdd(&s_conf[slot], conf);
    atomicAdd(&s_cnt [slot], 1.0f);
    atomicAdd(&s_acc [slot], labels[i]);
  }
  __syncthreads();

  // Wave 0 (fully active, EXEC all 1s) reduces the three 16x32 tiles with WMMA.
  if (tid < 32) {
    const v8f dc = lds_colsum_16x32(s_conf, tid);
    const v8f dn = lds_colsum_16x32(s_cnt,  tid);
    const v8f da = lds_colsum_16x32(s_acc,  tid);
    if (tid < NBINS) {  // D[0][n] lives in VGPR0 of lanes 0..15
      atomicAdd(&ws[tid],      dn[0]);
      atomicAdd(&ws[16 + tid], dc[0]);
      atomicAdd(&ws[32 + tid], da[0]);
    }
  }
}

__global__ void __launch_bounds__(32)
ece_final(const float* __restrict__ ws, float* __restrict__ out, float inv_n) {
  if (threadIdx.x == 0) {
    float ece = 0.0f;
#pragma unroll
    for (int b = 0; b < NBINS; ++b) {
      const float cnt = ws[b];
      if (cnt > 0.0f) {
        const float gap = fabsf((ws[16 + b] - ws[32 + b]) / cnt);  // |avg_conf - acc|
        ece += gap * cnt * inv_n;                                   // * proportion
      }
    }
    out[0] = ece;
  }
}

extern "C" void kernel_launch(void* const* d_in, const int* in_sizes, int n_in,
                              void* d_out, int out_size, void* d_ws, size_t ws_size,
                              hipStream_t stream) {
  const float* logits = (const float*)d_in[0];
  const float* labels = (const float*)d_in[1];
  float* out = (float*)d_out;
  float* ws  = (float*)d_ws;
  const int n = in_sizes[0];

  ece_zero_ws<<<1, 64, 0, stream>>>(ws);

  int blocks = (n + 256 * 4 - 1) / (256 * 4);
  if (blocks > 2048) blocks = 2048;  // ~32 grid-stride sweeps at 64M elements
  if (blocks < 1)    blocks = 1;
  ece_hist<<<blocks, 256, 0, stream>>>(logits, labels, ws, n);

  ece_final<<<1, 32, 0, stream>>>(ws, out, 1.0f / (float)n);
}